// DualEncoder_15650860827134
// MI455X (gfx1250) — compile-verified
//
#include <hip/hip_runtime.h>
#include <math.h>

// D = A x B + C ; f32 WMMA on gfx1250: A=16x4 (v2f), B=4x16 (v2f), C/D=16x16 (v8f)
typedef __attribute__((ext_vector_type(2))) float v2f;
typedef __attribute__((ext_vector_type(8))) float v8f;

#define BATCH  999
#define SEQ    160
#define HID    1024
#define TILE_B 16
#define NWAVES 8
#define LDW    (HID + 4)   // +4 pad: row stride 4112B -> 16 lanes hit 16 distinct banks

__global__ __launch_bounds__(256)
void DualEncoder_bilinear_wmma(const float* __restrict__ ctx,
                               const float* __restrict__ rsp,
                               const float* __restrict__ M,
                               float* __restrict__ out)
{
    __shared__ float sC[TILE_B * LDW];        // 16 x 1024 C-tile (last tokens), ~64.3 KB
    __shared__ float sPart[NWAVES][TILE_B];   // per-wave partial dots

    const int tid  = threadIdx.x;
    const int lane = tid & 31;
    const int wave = tid >> 5;
    const int b0   = blockIdx.x * TILE_B;

    // ---- Stage C tile (context last-token rows) into LDS, float4 per thread ----
    // 4096 float4 total / 256 threads = 16 iters, uniform trip count (EXEC stays full).
    for (int i = tid; i < TILE_B * (HID / 4); i += 256) {
        const int row = i >> 8;        // / (HID/4)
        const int c4  = i & 255;
        int gb = b0 + row; if (gb > BATCH - 1) gb = BATCH - 1;   // clamp tail tile
        const size_t goff = (size_t)gb * (SEQ * HID) + (size_t)(SEQ - 1) * HID + (size_t)c4 * 4;
        *reinterpret_cast<float4*>(&sC[row * LDW + c4 * 4]) =
            *reinterpret_cast<const float4*>(ctx + goff);
    }
    __syncthreads();

    // ISA operand layouts (wave32):
    //  A 16x4 f32 : lanes 0-15 -> M=lane, {K,K+1}; lanes 16-31 -> M=lane-16, {K+2,K+3}
    //  B 4x16 f32 : mirrored -> lanes 0-15 N=lane {K,K+1}; lanes 16-31 N=lane-16 {K+2,K+3}
    //  C/D 16x16  : lane = column (mod 16); VGPR j -> row j (lanes 0-15) / row j+8 (16-31)
    const int col   = lane & 15;
    const int khalf = (lane >> 4) * 2;
    const int mbase = (lane >> 4) * 8;

    // r-row pointers for the 8 accumulator rows this lane sees (no reuse -> global).
    const float* rRow[8];
    #pragma unroll
    for (int j = 0; j < 8; ++j) {
        int gb = b0 + mbase + j; if (gb > BATCH - 1) gb = BATCH - 1;
        rRow[j] = rsp + (size_t)gb * (SEQ * HID) + (size_t)(SEQ - 1) * HID;
    }

    const float* __restrict__ aRow = &sC[col * LDW + khalf];

    float dotp[8];
    #pragma unroll
    for (int j = 0; j < 8; ++j) dotp[j] = 0.f;

    // ---- Each wave: N-chunks of 16 columns, strided by NWAVES ----
    for (int nc = wave; nc < HID / 16; nc += NWAVES) {
        const int n0 = nc * 16;
        const float* __restrict__ bPtr = M + (size_t)khalf * HID + n0 + col;

        v8f acc = {};
        #pragma unroll 4
        for (int k = 0; k < HID; k += 4) {
            v2f a, b;
            a.x = aRow[k];
            a.y = aRow[k + 1];
            b.x = bPtr[(size_t)k * HID];
            b.y = bPtr[(size_t)(k + 1) * HID];
            acc = __builtin_amdgcn_wmma_f32_16x16x4_f32(
                /*neg_a=*/false, a, /*neg_b=*/false, b,
                /*c_mod=*/(short)0, acc, /*reuse_a=*/false, /*reuse_b=*/false);
        }

        // Fold T[b, n0+col] into the per-batch dot against r immediately.
        #pragma unroll
        for (int j = 0; j < 8; ++j)
            dotp[j] += acc[j] * rRow[j][n0 + col];
    }

    // ---- Reduce the 16 column-lanes of each half-wave (xor-shuffle tree) ----
    #pragma unroll
    for (int j = 0; j < 8; ++j) {
        float v = dotp[j];
        v += __shfl_xor(v, 1);
        v += __shfl_xor(v, 2);
        v += __shfl_xor(v, 4);
        v += __shfl_xor(v, 8);
        if (col == 0) sPart[wave][mbase + j] = v;   // lane0 -> rows 0-7, lane16 -> rows 8-15
    }
    __syncthreads();

    // ---- Fixed-order cross-wave sum (deterministic, no float atomics) + sigmoid ----
    if (tid < TILE_B) {
        float s = 0.f;
        #pragma unroll
        for (int w = 0; w < NWAVES; ++w) s += sPart[w][tid];
        const int gb = b0 + tid;
        if (gb < BATCH) out[gb] = 1.0f / (1.0f + expf(-s));
    }
}

extern "C" void kernel_launch(void* const* d_in, const int* in_sizes, int n_in,
                              void* d_out, int out_size, void* d_ws, size_t ws_size,
                              hipStream_t stream) {
    const float* ctx = (const float*)d_in[0];   // context_out: (999,160,1024) f32
    const float* rsp = (const float*)d_in[1];   // response_out: (999,160,1024) f32
    const float* M   = (const float*)d_in[2];   // M: (1024,1024) f32
    float* out = (float*)d_out;                 // (999,) f32

    const int grid = (BATCH + TILE_B - 1) / TILE_B;   // 63 batch tiles
    DualEncoder_bilinear_wmma<<<grid, 256, 0, stream>>>(ctx, rsp, M, out);
}